// DFAChebNet_7876970020889
// MI455X (gfx1250) — compile-verified
//
#include <hip/hip_runtime.h>
#include <math.h>

#define NNODES 100000
#define NEDGES 1600000
#define FIN 128
#define HID 16
#define NCLS 32
#define TILES ((NNODES + 15) / 16) // 6250, exact

typedef __attribute__((ext_vector_type(2))) float v2f;
typedef __attribute__((ext_vector_type(4))) float v4f;
typedef __attribute__((ext_vector_type(8))) float v8f;

// ---------------- prep kernels ----------------

__global__ void deg_kernel(const long long* __restrict__ row,
                           const float* __restrict__ w,
                           float* __restrict__ deg) {
    int e = blockIdx.x * blockDim.x + threadIdx.x;
    if (e < NEDGES) atomicAdd(&deg[(int)row[e]], w[e]);
}

__global__ void dinv_kernel(float* __restrict__ deg) {
    int i = blockIdx.x * blockDim.x + threadIdx.x;
    if (i < NNODES) {
        float d = deg[i];
        deg[i] = d > 0.f ? rsqrtf(d) : 0.f;
    }
}

__global__ void norm_kernel(const long long* __restrict__ row,
                            const long long* __restrict__ col,
                            const float* __restrict__ w,
                            const float* __restrict__ dinv,
                            float* __restrict__ nrm) {
    int e = blockIdx.x * blockDim.x + threadIdx.x;
    if (e < NEDGES) nrm[e] = dinv[(int)row[e]] * w[e] * dinv[(int)col[e]];
}

__global__ void wdiff_kernel(const float* __restrict__ W10, const float* __restrict__ W11,
                             const float* __restrict__ W20, const float* __restrict__ W21,
                             float* __restrict__ Wd1, float* __restrict__ Wd2) {
    int i = blockIdx.x * blockDim.x + threadIdx.x;
    if (i < FIN * HID)  Wd1[i] = W10[i] - W11[i];
    if (i < HID * NCLS) Wd2[i] = W20[i] - W21[i];
}

// ---------------- layer-1 GEMM: y0 = x@(W0-W1)+b1 (-> agg1), y1 = x@W1 ----------------
// One wave per 16-row tile. fp32 WMMA 16x16x4, K=128 -> 32 steps, 2 accumulators.

__global__ __launch_bounds__(256) void gemm1_kernel(
        const float* __restrict__ x,
        const float* __restrict__ Wd1,  // 128x16
        const float* __restrict__ W11,  // 128x16
        const float* __restrict__ b1,
        float* __restrict__ agg1,       // N x 16 : y0 + b1
        float* __restrict__ y1) {       // N x 16
    __shared__ float sWd[FIN * HID];
    __shared__ float sW1[FIN * HID];
    for (int i = threadIdx.x; i < FIN * HID; i += blockDim.x) {
        sWd[i] = Wd1[i];
        sW1[i] = W11[i];
    }
    __syncthreads();

    int wave = threadIdx.x >> 5;
    int lane = threadIdx.x & 31;
    int tile = blockIdx.x * (blockDim.x >> 5) + wave;
    if (tile >= TILES) return;

    int m = lane & 15;          // A row / B column index for this lane
    int half = lane >> 4;       // K-pair selector
    const float* arow = x + (size_t)(tile * 16 + m) * FIN;

    v8f c0 = {}; // x @ (W0-W1)
    v8f c1 = {}; // x @ W1
#pragma unroll 4
    for (int k = 0; k < FIN; k += 4) {
        int k0 = k + 2 * half;
        v2f a = *(const v2f*)(arow + k0);           // A[m][k0], A[m][k0+1]
        v2f bd, bw;
        bd.x = sWd[k0 * HID + m]; bd.y = sWd[(k0 + 1) * HID + m];
        bw.x = sW1[k0 * HID + m]; bw.y = sW1[(k0 + 1) * HID + m];
        c0 = __builtin_amdgcn_wmma_f32_16x16x4_f32(false, a, false, bd, (short)0, c0, false, false);
        c1 = __builtin_amdgcn_wmma_f32_16x16x4_f32(false, a, false, bw, (short)0, c1, false, false);
    }

    float bias = b1[m];
    int base = tile * 16;
#pragma unroll
    for (int r = 0; r < 8; r++) {
        int rr = base + r + 8 * half;               // D: VGPR r holds row r (+8 for upper half)
        agg1[rr * HID + m] = c0[r] + bias;
        y1[rr * HID + m]   = c1[r];
    }
}

// ------- edge scatter, layer 1: agg1[row] += norm * y1[col]; 4 feats/thread, float4 gather

__global__ void scatter1_kernel(const long long* __restrict__ row,
                                const long long* __restrict__ col,
                                const float* __restrict__ nrm,
                                const float* __restrict__ y1,
                                float* __restrict__ agg1) {
    int t = blockIdx.x * blockDim.x + threadIdx.x;
    int e = t >> 2;            // 4 threads per edge
    int fq = t & 3;            // feature quad
    if (e < NEDGES) {
        int r = (int)row[e];
        int c = (int)col[e];
        float s = nrm[e];
        v4f v = *(const v4f*)(y1 + c * HID + fq * 4);
        float* dst = agg1 + r * HID + fq * 4;
#pragma unroll
        for (int j = 0; j < 4; j++) atomicAdd(dst + j, s * v[j]);
    }
}

// ---------------- layer-2 GEMM: h = relu(agg1); z0 = h@(W0-W1)+b2 (-> out), z1 = h@W1 ----

__global__ __launch_bounds__(256) void gemm2_kernel(
        const float* __restrict__ agg1, // pre-relu hidden, N x 16
        const float* __restrict__ Wd2,  // 16x32
        const float* __restrict__ W21,  // 16x32
        const float* __restrict__ b2,
        float* __restrict__ out,        // N x 32 : z0 + b2
        float* __restrict__ z1) {       // N x 32
    __shared__ float sWd[HID * NCLS];
    __shared__ float sW2[HID * NCLS];
    for (int i = threadIdx.x; i < HID * NCLS; i += blockDim.x) {
        sWd[i] = Wd2[i];
        sW2[i] = W21[i];
    }
    __syncthreads();

    int wave = threadIdx.x >> 5;
    int lane = threadIdx.x & 31;
    int tile = blockIdx.x * (blockDim.x >> 5) + wave;
    if (tile >= TILES) return;

    int m = lane & 15;
    int half = lane >> 4;
    const float* arow = agg1 + (size_t)(tile * 16 + m) * HID;

    v8f c00 = {}, c01 = {}; // z0, N-halves 0 / 1
    v8f c10 = {}, c11 = {}; // z1, N-halves 0 / 1
#pragma unroll
    for (int k = 0; k < HID; k += 4) {
        int k0 = k + 2 * half;
        v2f a;
        a.x = fmaxf(arow[k0], 0.f);                 // fused ReLU
        a.y = fmaxf(arow[k0 + 1], 0.f);
        v2f bd0, bd1, bw0, bw1;
        bd0.x = sWd[k0 * NCLS + m];        bd0.y = sWd[(k0 + 1) * NCLS + m];
        bd1.x = sWd[k0 * NCLS + 16 + m];   bd1.y = sWd[(k0 + 1) * NCLS + 16 + m];
        bw0.x = sW2[k0 * NCLS + m];        bw0.y = sW2[(k0 + 1) * NCLS + m];
        bw1.x = sW2[k0 * NCLS + 16 + m];   bw1.y = sW2[(k0 + 1) * NCLS + 16 + m];
        c00 = __builtin_amdgcn_wmma_f32_16x16x4_f32(false, a, false, bd0, (short)0, c00, false, false);
        c01 = __builtin_amdgcn_wmma_f32_16x16x4_f32(false, a, false, bd1, (short)0, c01, false, false);
        c10 = __builtin_amdgcn_wmma_f32_16x16x4_f32(false, a, false, bw0, (short)0, c10, false, false);
        c11 = __builtin_amdgcn_wmma_f32_16x16x4_f32(false, a, false, bw1, (short)0, c11, false, false);
    }

    float bias0 = b2[m];
    float bias1 = b2[16 + m];
    int base = tile * 16;
#pragma unroll
    for (int r = 0; r < 8; r++) {
        int rr = base + r + 8 * half;
        out[rr * NCLS + m]      = c00[r] + bias0;
        out[rr * NCLS + 16 + m] = c01[r] + bias1;
        z1[rr * NCLS + m]       = c10[r];
        z1[rr * NCLS + 16 + m]  = c11[r];
    }
}

// ------- edge scatter, layer 2: out[row] += norm * z1[col]; 4 feats/thread, float4 gather

__global__ void scatter2_kernel(const long long* __restrict__ row,
                                const long long* __restrict__ col,
                                const float* __restrict__ nrm,
                                const float* __restrict__ z1,
                                float* __restrict__ out) {
    int t = blockIdx.x * blockDim.x + threadIdx.x;
    int e = t >> 3;            // 8 threads per edge
    int fq = t & 7;            // feature quad
    if (e < NEDGES) {
        int r = (int)row[e];
        int c = (int)col[e];
        float s = nrm[e];
        v4f v = *(const v4f*)(z1 + c * NCLS + fq * 4);
        float* dst = out + r * NCLS + fq * 4;
#pragma unroll
        for (int j = 0; j < 4; j++) atomicAdd(dst + j, s * v[j]);
    }
}

// ---------------- in-place log_softmax: one wave32 per row, one lane per class ----------

__global__ void logsoftmax_kernel(float* __restrict__ out) {
    int wave = threadIdx.x >> 5;
    int lane = threadIdx.x & 31;
    int i = blockIdx.x * (blockDim.x >> 5) + wave;
    if (i >= NNODES) return;
    float v = out[i * NCLS + lane];
    float mx = v;
#pragma unroll
    for (int off = 16; off >= 1; off >>= 1)
        mx = fmaxf(mx, __shfl_xor(mx, off, 32));
    float ex = expf(v - mx);
    float s = ex;
#pragma unroll
    for (int off = 16; off >= 1; off >>= 1)
        s += __shfl_xor(s, off, 32);
    out[i * NCLS + lane] = v - mx - logf(s);
}

// ---------------- host launch ----------------

extern "C" void kernel_launch(void* const* d_in, const int* in_sizes, int n_in,
                              void* d_out, int out_size, void* d_ws, size_t ws_size,
                              hipStream_t stream) {
    const float*     x   = (const float*)d_in[0];
    const long long* ei  = (const long long*)d_in[1];  // int64 [2, E]
    const float*     w   = (const float*)d_in[2];
    const float*     W10 = (const float*)d_in[3];
    const float*     W11 = (const float*)d_in[4];
    const float*     b1  = (const float*)d_in[5];
    const float*     W20 = (const float*)d_in[6];
    const float*     W21 = (const float*)d_in[7];
    const float*     b2  = (const float*)d_in[8];
    float* out = (float*)d_out;

    const long long* row = ei;
    const long long* col = ei + NEDGES;

    float* ws   = (float*)d_ws;
    float* deg  = ws;  ws += NNODES;         // then holds dinv in-place
    float* nrm  = ws;  ws += NEDGES;
    float* y1   = ws;  ws += (size_t)NNODES * HID;
    float* agg1 = ws;  ws += (size_t)NNODES * HID;
    float* z1   = ws;  ws += (size_t)NNODES * NCLS;
    float* Wd1  = ws;  ws += FIN * HID;
    float* Wd2  = ws;  ws += HID * NCLS;

    hipMemsetAsync(deg, 0, NNODES * sizeof(float), stream);

    deg_kernel <<<(NEDGES + 255) / 256, 256, 0, stream>>>(row, w, deg);
    dinv_kernel<<<(NNODES + 255) / 256, 256, 0, stream>>>(deg);
    norm_kernel<<<(NEDGES + 255) / 256, 256, 0, stream>>>(row, col, w, deg, nrm);
    wdiff_kernel<<<(FIN * HID + 255) / 256, 256, 0, stream>>>(W10, W11, W20, W21, Wd1, Wd2);

    gemm1_kernel<<<(TILES + 7) / 8, 256, 0, stream>>>(x, Wd1, W11, b1, agg1, y1);
    scatter1_kernel<<<(NEDGES * 4 + 255) / 256, 256, 0, stream>>>(row, col, nrm, y1, agg1);

    gemm2_kernel<<<(TILES + 7) / 8, 256, 0, stream>>>(agg1, Wd2, W21, b2, out, z1);
    scatter2_kernel<<<(NEDGES * 8 + 255) / 256, 256, 0, stream>>>(row, col, nrm, z1, out);

    logsoftmax_kernel<<<(NNODES + 7) / 8, 256, 0, stream>>>(out);
}